// EmbeddingNet_Pos_6511170421156
// MI455X (gfx1250) — compile-verified
//
#include <hip/hip_runtime.h>
#include <stdint.h>

// Shapes fixed by the reference: B=128, S=2048, D=128.
#define BDIM 128
#define SDIM 2048
#define DDIM 128

typedef float v4f __attribute__((ext_vector_type(4)));
typedef int   v4i __attribute__((ext_vector_type(4)));

// pe[r][j]: angle = (r+1) / 10000^(2*(j/2)/128); row 0 stays raw angle,
// rows >=1 get sin on even j, cos on odd j (faithful to the reference).
__device__ __forceinline__ float pe_val(int r, int j) {
    float e     = (float)(2 * (j >> 1)) * (1.0f / (float)DDIM);
    float angle = (float)(r + 1) * powf(10000.0f, -e);
    if (r == 0) return angle;
    return (j & 1) ? cosf(angle) : sinf(angle);
}

// Kernel A: build the [S, D] PE table in workspace (1 MB, one thread per elem).
__global__ void pe_init_kernel(float* __restrict__ pe) {
    int t = blockIdx.x * blockDim.x + threadIdx.x;   // 0 .. S*D-1
    int r = t >> 7;                                   // D == 128
    int j = t & (DDIM - 1);
    pe[t] = pe_val(r, j);
}

// Kernel B: scatter PE rows. argsort(perm) == inverse perm, so
//   out[which][b, perm[i], :] = pe[i, :].
// One wave handles 4 consecutive rows (same b, same output tensor):
//   - 4 async global->LDS B128 loads (512 B/row), one s_wait_asynccnt
//   - ds_load_b128 readback
//   - non-temporal global_store_b128 to the scattered destination row.
template <bool USE_TABLE>
__global__ void scatter_kernel(const int* __restrict__ sol,
                               const int* __restrict__ best,
                               const float* __restrict__ pe,
                               float* __restrict__ out) {
    extern __shared__ char lds_raw[];   // 16 KB dynamic when USE_TABLE
    (void)lds_raw;

    const unsigned tid  = blockIdx.x * blockDim.x + threadIdx.x;
    const unsigned lane = threadIdx.x & 31u;
    const unsigned w    = tid >> 5;          // global wave id
    const unsigned r0   = w << 2;            // first of 4 rows for this wave
    const unsigned which = r0 >> 18;         // rows per output = B*S = 2^18
    const unsigned rem   = r0 & 0x3FFFFu;
    const unsigned b     = rem >> 11;        // S = 2^11
    const unsigned i0    = rem & (SDIM - 1); // multiple of 4 -> i0..i0+3 same b

    const int* sp = which ? best : sol;
    // 4 destination rows, read once -> non-temporal 16B load
    v4i dstRows = __builtin_nontemporal_load(
        (const v4i*)(sp + (size_t)b * SDIM + i0));

    v4f vals[4];
    if (USE_TABLE) {
        const unsigned wib  = threadIdx.x >> 5;   // wave-in-block 0..7
        const unsigned slot = wib * 2048u;        // 4 rows * 512 B per wave
        #pragma unroll
        for (int k = 0; k < 4; ++k) {
            unsigned lds_off = slot + (unsigned)k * 512u + lane * 16u;
            const float* src = pe + (((size_t)i0 + (unsigned)k) << 7) + (lane << 2);
            unsigned long long ga = (unsigned long long)(uintptr_t)src;
            asm volatile("global_load_async_to_lds_b128 %0, %1, off"
                         :: "v"(lds_off), "v"(ga) : "memory");
        }
        asm volatile("s_wait_asynccnt 0" ::: "memory");
        #pragma unroll
        for (int k = 0; k < 4; ++k) {
            unsigned lds_off = slot + (unsigned)k * 512u + lane * 16u;
            v4f v;
            asm volatile("ds_load_b128 %0, %1\n\t"
                         "s_wait_dscnt 0"
                         : "=v"(v) : "v"(lds_off) : "memory");
            vals[k] = v;
        }
    } else {
        // Workspace-too-small fallback: compute PE inline (still correct).
        const int j0 = (int)(lane << 2);
        #pragma unroll
        for (int k = 0; k < 4; ++k) {
            v4f v;
            #pragma unroll
            for (int q = 0; q < 4; ++q) v[q] = pe_val((int)i0 + k, j0 + q);
            vals[k] = v;
        }
    }

    // out[which][b, dstRow, :] ; which<<25 = B*S*D, b<<18 = S*D
    float* outBase = out + ((size_t)which << 25) + ((size_t)b << 18);
    #pragma unroll
    for (int k = 0; k < 4; ++k) {
        int dr = dstRows[k];                               // in [0, S)
        float* dst = outBase + ((size_t)dr << 7) + (lane << 2);
        __builtin_nontemporal_store(vals[k], (v4f*)dst);   // 256MB stream: NT
    }
}

extern "C" void kernel_launch(void* const* d_in, const int* in_sizes, int n_in,
                              void* d_out, int out_size, void* d_ws, size_t ws_size,
                              hipStream_t stream) {
    (void)in_sizes; (void)n_in; (void)out_size;
    // d_in[0] = x (unused by the reference outputs)
    const int* sol  = (const int*)d_in[1];
    const int* best = (const int*)d_in[2];
    float* out = (float*)d_out;

    const size_t PE_BYTES = (size_t)SDIM * DDIM * sizeof(float);  // 1 MB

    // total rows = 2 * B * S = 524288; 4 rows/wave, 8 waves/block
    // -> 131072 waves -> 4,194,304 threads -> 16384 blocks of 256
    const int blocksB = (2 * BDIM * SDIM / 4) / 8;

    if (ws_size >= PE_BYTES) {
        float* pe = (float*)d_ws;
        pe_init_kernel<<<(SDIM * DDIM) / 256, 256, 0, stream>>>(pe);
        scatter_kernel<true><<<blocksB, 256, 16384, stream>>>(sol, best, pe, out);
    } else {
        scatter_kernel<false><<<blocksB, 256, 0, stream>>>(sol, best, nullptr, out);
    }
}